// SEM_42236708389411
// MI455X (gfx1250) — compile-verified
//
#include <hip/hip_runtime.h>
#include <math.h>

// ---------------- problem constants ----------------
#define BB 64
#define NN 1024
#define DD 128
#define RR 640
#define KK 32
#define LL 80

// d_out layout: out[B,K,D] | policy[B] | scorer[B,D] | entropy[B]
#define O_OUT    0
#define O_POL    (BB*KK*DD)            // 262144
#define O_SCORER (O_POL + BB)          // 262208
#define O_ENT    (O_SCORER + BB*DD)    // 270400

// workspace layout (floats) ~6.7 MiB
#define W_TANH   0                      // B*D
#define W_NORM   (W_TANH + BB*DD)       // B
#define W_SCORES (W_NORM + BB)          // B*N
#define W_TKV    (W_SCORES + BB*NN)     // B*K
#define W_TKI    (W_TKV + BB*KK)        // B*K (ints)
#define W_NSEL   (W_TKI + BB*KK)        // B*K*D
#define W_SGATE  (W_NSEL + BB*KK*DD)    // B*K*D
#define W_WLF    (W_SGATE + BB*KK*DD)   // B*L*D
#define W_ATTN   (W_WLF + BB*LL*DD)     // B*K*L
#define W_AF     (W_ATTN + BB*KK*LL)    // B*K*K
#define W_BASE   (W_AF + BB*KK*KK)      // B*K*D

typedef __attribute__((ext_vector_type(2))) float v2f;
typedef __attribute__((ext_vector_type(8))) float v8f;

// ---- generic fp32 WMMA 16x16 tile accumulators (V_WMMA_F32_16X16X4_F32) ----
// A: 16 x ksz row-major (lda).  B: ksz x 16 row-major (ldb).
__device__ __forceinline__ v8f wmma_acc_rmaj(const float* A, int lda,
                                             const float* B, int ldb,
                                             int ksz, v8f acc) {
  const int lane = threadIdx.x & 31;
  const int half = lane >> 4, l16 = lane & 15;
  for (int k0 = 0; k0 < ksz; k0 += 4) {
    const int ka = k0 + 2 * half;
    v2f a, b;
    a[0] = A[l16 * lda + ka];
    a[1] = A[l16 * lda + ka + 1];
    b[0] = B[ka * ldb + l16];
    b[1] = B[(ka + 1) * ldb + l16];
    acc = __builtin_amdgcn_wmma_f32_16x16x4_f32(false, a, false, b,
                                                (short)0, acc, false, false);
  }
  return acc;
}

// B given transposed: BT is 16 x ksz row-major (ldbt), i.e. B[k][n] = BT[n*ldbt+k]
__device__ __forceinline__ v8f wmma_acc_btrans(const float* A, int lda,
                                               const float* BT, int ldbt,
                                               int ksz, v8f acc) {
  const int lane = threadIdx.x & 31;
  const int half = lane >> 4, l16 = lane & 15;
  for (int k0 = 0; k0 < ksz; k0 += 4) {
    const int ka = k0 + 2 * half;
    v2f a, b;
    a[0] = A[l16 * lda + ka];
    a[1] = A[l16 * lda + ka + 1];
    b[0] = BT[l16 * ldbt + ka];
    b[1] = BT[l16 * ldbt + ka + 1];
    acc = __builtin_amdgcn_wmma_f32_16x16x4_f32(false, a, false, b,
                                                (short)0, acc, false, false);
  }
  return acc;
}

// ------------- K1: scorer = tanh(ht @ W_topk^T + b); tanh_init likewise -----
__global__ void k_scorer(const float* __restrict__ ht,
                         const float* __restrict__ Wt, const float* __restrict__ bt,
                         const float* __restrict__ Wi, const float* __restrict__ bi,
                         float* __restrict__ scorer, float* __restrict__ tanh_init) {
  const int tm = blockIdx.x;   // 4 tiles of B=64 rows
  const int tn = blockIdx.y;   // 8 tiles of D=128 cols
  const int which = blockIdx.z;
  const float* W = which ? Wi : Wt;
  const float* bias = which ? bi : bt;
  float* dst = which ? tanh_init : scorer;
  v8f acc = {};
  acc = wmma_acc_btrans(ht + tm * 16 * RR, RR, W + tn * 16 * RR, RR, RR, acc);
  const int lane = threadIdx.x & 31, half = lane >> 4, l16 = lane & 15;
  const float bb = bias[tn * 16 + l16];
#pragma unroll
  for (int v = 0; v < 8; v++) {
    dst[(tm * 16 + v + 8 * half) * DD + tn * 16 + l16] = tanhf(acc[v] + bb);
  }
}

// ------------- K1b: per-batch scorer L2 norm -------------------------------
__global__ void k_norm(const float* __restrict__ scorer, float* __restrict__ norms) {
  const int b = blockIdx.x, lane = threadIdx.x;
  float ss = 0.f;
  for (int d = lane; d < DD; d += 32) { float v = scorer[b * DD + d]; ss += v * v; }
#pragma unroll
  for (int m = 16; m > 0; m >>= 1) ss += __shfl_xor(ss, m, 32);
  if (lane == 0) norms[b] = sqrtf(ss);
}

// ------------- K2: scores = (node_embs . scorer)/norm, masked -> 1e-9 ------
__global__ void k_scores(const float* __restrict__ node_embs,
                         const float* __restrict__ scorer,
                         const float* __restrict__ norms,
                         const int* __restrict__ mask,
                         float* __restrict__ scores) {
  const int wid = (int)((blockIdx.x * blockDim.x + threadIdx.x) >> 5); // b*N+n
  const int lane = threadIdx.x & 31;
  const int b = wid >> 10;
  const float* e = node_embs + (size_t)wid * DD;
  const float* sc = scorer + b * DD;
  float dot = 0.f;
  for (int d = lane; d < DD; d += 32) dot += e[d] * sc[d];
#pragma unroll
  for (int m = 16; m > 0; m >>= 1) dot += __shfl_xor(dot, m, 32);
  if (lane == 0) {
    float v = dot / norms[b];
    if (mask[wid] != 0) v = 1e-9f;
    scores[wid] = v;
  }
}

// ------------- K3: per-batch log-softmax stats, entropy, top-32, policy ----
__global__ void k_topk(const float* __restrict__ scores,
                       float* __restrict__ policy, float* __restrict__ entropy,
                       float* __restrict__ tkv, int* __restrict__ tki) {
  __shared__ float sv[NN];
  __shared__ float red[256];
  __shared__ int redi[256];
  __shared__ float s_m, s_lse;
  const int b = blockIdx.x, tid = threadIdx.x;
  for (int i = tid; i < NN; i += 256) sv[i] = scores[b * NN + i];
  __syncthreads();
  // max
  float m = -INFINITY;
  for (int i = tid; i < NN; i += 256) m = fmaxf(m, sv[i]);
  red[tid] = m; __syncthreads();
  for (int s = 128; s > 0; s >>= 1) {
    if (tid < s) red[tid] = fmaxf(red[tid], red[tid + s]);
    __syncthreads();
  }
  if (tid == 0) s_m = red[0];
  __syncthreads();
  const float mm = s_m;
  // log-sum-exp
  float se = 0.f;
  for (int i = tid; i < NN; i += 256) se += expf(sv[i] - mm);
  red[tid] = se; __syncthreads();
  for (int s = 128; s > 0; s >>= 1) {
    if (tid < s) red[tid] += red[tid + s];
    __syncthreads();
  }
  if (tid == 0) s_lse = mm + logf(red[0]);
  __syncthreads();
  const float lse = s_lse;
  // entropy = -sum p*logp
  float ent = 0.f;
  for (int i = tid; i < NN; i += 256) { float lp = sv[i] - lse; ent += expf(lp) * lp; }
  red[tid] = ent; __syncthreads();
  for (int s = 128; s > 0; s >>= 1) {
    if (tid < s) red[tid] += red[tid + s];
    __syncthreads();
  }
  if (tid == 0) entropy[b] = -red[0];
  __syncthreads();
  // iterative top-32 (value desc, index asc on ties, matching jax.lax.top_k)
  float psum = 0.f;
  for (int kk = 0; kk < KK; kk++) {
    float bv = -INFINITY; int bi = 1 << 30;
    for (int i = tid; i < NN; i += 256) {
      float v = sv[i];
      if (v > bv || (v == bv && i < bi)) { bv = v; bi = i; }
    }
    red[tid] = bv; redi[tid] = bi; __syncthreads();
    for (int s = 128; s > 0; s >>= 1) {
      if (tid < s) {
        float v2 = red[tid + s]; int i2 = redi[tid + s];
        if (v2 > red[tid] || (v2 == red[tid] && i2 < redi[tid])) {
          red[tid] = v2; redi[tid] = i2;
        }
      }
      __syncthreads();
    }
    if (tid == 0) {
      int bidx = redi[0]; float bval = red[0];
      tki[b * KK + kk] = bidx;
      tkv[b * KK + kk] = bval;
      sv[bidx] = -INFINITY;
      psum += bval - lse;
    }
    __syncthreads();
  }
  if (tid == 0) policy[b] = psum * (1.0f / KK);
}

// ------------- K4: gather node_sel + sel_gate ------------------------------
__global__ void k_gather(const float* __restrict__ node_embs,
                         const int* __restrict__ tki, const float* __restrict__ tkv,
                         float* __restrict__ node_sel, float* __restrict__ sel_gate) {
  const int bk = blockIdx.x;        // b*K + k
  const int b = bk >> 5;
  const int idx = tki[bk];
  const float g = tanhf(tkv[bk]);
  const float* src = node_embs + ((size_t)b * NN + idx) * DD;
  const int d = threadIdx.x;
  const float v = src[d];
  node_sel[(size_t)bk * DD + d] = v;
  sel_gate[(size_t)bk * DD + d] = v * g;
}

// ------------- K5: wlf = word_feats @ W_map^T + b_map (WMMA) ---------------
__global__ void k_wlf(const float* __restrict__ wf, const float* __restrict__ Wmap,
                      const float* __restrict__ bmap, float* __restrict__ wlf) {
  const int tm = blockIdx.x, tn = blockIdx.y, b = blockIdx.z;
  v8f acc = {};
  acc = wmma_acc_btrans(wf + (size_t)b * LL * RR + tm * 16 * RR, RR,
                        Wmap + tn * 16 * RR, RR, RR, acc);
  const int lane = threadIdx.x & 31, half = lane >> 4, l16 = lane & 15;
  const float bb = bmap[tn * 16 + l16];
  float* dst = wlf + (size_t)b * LL * DD;
#pragma unroll
  for (int v = 0; v < 8; v++)
    dst[(tm * 16 + v + 8 * half) * DD + tn * 16 + l16] = acc[v] + bb;
}

// ------------- K6: attn = node_sel @ wlf^T (WMMA) --------------------------
__global__ void k_attn(const float* __restrict__ node_sel,
                       const float* __restrict__ wlf, float* __restrict__ attn) {
  const int tm = blockIdx.x, tn = blockIdx.y, b = blockIdx.z;
  v8f acc = {};
  acc = wmma_acc_btrans(node_sel + (size_t)b * KK * DD + tm * 16 * DD, DD,
                        wlf + (size_t)b * LL * DD + tn * 16 * DD, DD, DD, acc);
  const int lane = threadIdx.x & 31, half = lane >> 4, l16 = lane & 15;
  float* dst = attn + (size_t)b * KK * LL;
#pragma unroll
  for (int v = 0; v < 8; v++)
    dst[(tm * 16 + v + 8 * half) * LL + tn * 16 + l16] = acc[v];
}

// ------------- K7: GRU cell + attn@attn^T + adj sym-norm -> A_final --------
__global__ void k_afinal(const float* __restrict__ Adj, const float* __restrict__ evA,
                         const float* __restrict__ Wz, const float* __restrict__ Uz, const float* __restrict__ bz,
                         const float* __restrict__ Wr, const float* __restrict__ Ur, const float* __restrict__ br,
                         const float* __restrict__ Wh, const float* __restrict__ Uh, const float* __restrict__ bh,
                         const float* __restrict__ sel_gate, const float* __restrict__ attn,
                         const int* __restrict__ tki, float* __restrict__ A_final) {
  __shared__ float s_r[KK * KK];
  __shared__ float s_adj[KK * KK];
  __shared__ float s_di[KK];
  const int b = blockIdx.x, tid = threadIdx.x;
  const int lane = tid & 31, wv = tid >> 5;
  const int tm = wv >> 1, tn = wv & 1;        // 4 waves = 2x2 tiles of 32x32
  const int* idx = tki + b * KK;
  // gather adj_kk
  for (int t = tid; t < KK * KK; t += 128) {
    const int i = t >> 5, j = t & 31;
    s_adj[t] = Adj[(size_t)b * NN * NN + (size_t)idx[i] * NN + idx[j]];
  }
  const float* selg = sel_gate + (size_t)b * KK * DD;
  const float* Ab = evA + b * KK * KK;
  // z, r gates: W @ sel^T + U @ A + bias
  v8f zacc = {};
  zacc = wmma_acc_btrans(Wz + tm * 16 * DD, DD, selg + tn * 16 * DD, DD, DD, zacc);
  zacc = wmma_acc_rmaj(Uz + tm * 16 * KK, KK, Ab + tn * 16, KK, KK, zacc);
  v8f racc = {};
  racc = wmma_acc_btrans(Wr + tm * 16 * DD, DD, selg + tn * 16 * DD, DD, DD, racc);
  racc = wmma_acc_rmaj(Ur + tm * 16 * KK, KK, Ab + tn * 16, KK, KK, racc);
  const int half = lane >> 4, l16 = lane & 15;
  v8f zt;
#pragma unroll
  for (int v = 0; v < 8; v++) {
    const int i = tm * 16 + v + 8 * half, j = tn * 16 + l16;
    zt[v] = 1.f / (1.f + expf(-(zacc[v] + bz[i * KK + j])));
    s_r[i * KK + j] = 1.f / (1.f + expf(-(racc[v] + br[i * KK + j])));
  }
  __syncthreads();
  // rA in place; column sums for di (axis=1 sum)
  for (int t = tid; t < KK * KK; t += 128) s_r[t] *= Ab[t];
  if (tid < KK) {
    float s = 0.f;
    for (int i = 0; i < KK; i++) s += s_adj[i * KK + tid];
    s_di[tid] = 1.f / sqrtf(s);
  }
  __syncthreads();
  // h gate (U @ (r*A) from LDS) and attn_A = attn @ attn^T
  v8f hacc = {};
  hacc = wmma_acc_btrans(Wh + tm * 16 * DD, DD, selg + tn * 16 * DD, DD, DD, hacc);
  hacc = wmma_acc_rmaj(Uh + tm * 16 * KK, KK, s_r + tn * 16, KK, KK, hacc);
  v8f aacc = {};
  aacc = wmma_acc_btrans(attn + (size_t)b * KK * LL + tm * 16 * LL, LL,
                         attn + (size_t)b * KK * LL + tn * 16 * LL, LL, LL, aacc);
  float* dst = A_final + b * KK * KK;
#pragma unroll
  for (int v = 0; v < 8; v++) {
    const int i = tm * 16 + v + 8 * half, j = tn * 16 + l16;
    const float a = Ab[i * KK + j];
    const float h = tanhf(hacc[v] + bh[i * KK + j]);
    const float dnew = (1.f - zt[v]) * a + zt[v] * h;   // A_mid forward value
    const float adjn = s_di[i] * s_di[j] * s_adj[i * KK + j];
    dst[i * KK + j] = dnew + adjn + aacc[v];
  }
}

// ------------- K8: base[b,k,:] = (node_sel[b,k,:].tanh_init[b,:]) * g ------
__global__ void k_base(const float* __restrict__ node_sel,
                       const float* __restrict__ tanh_init,
                       const float* __restrict__ g, float* __restrict__ base) {
  const int bk = blockIdx.x;
  const int b = bk >> 5;
  const int lane = threadIdx.x;
  const float* ns = node_sel + (size_t)bk * DD;
  const float* t = tanh_init + b * DD;
  float s = 0.f;
  for (int d = lane; d < DD; d += 32) s += ns[d] * t[d];
#pragma unroll
  for (int m = 16; m > 0; m >>= 1) s += __shfl_xor(s, m, 32);
  float* dst = base + (size_t)bk * DD;
  for (int d = lane; d < DD; d += 32) dst[d] = s * g[d];
}

// ------------- K9: out = 0.5*(relu(A_final@base) + relu(A_static@base)) ----
__global__ void k_out(const float* __restrict__ A_final,
                      const float* __restrict__ A_static,
                      const float* __restrict__ base, float* __restrict__ out) {
  const int tm = blockIdx.x, tn = blockIdx.y, b = blockIdx.z;
  const float* Bb = base + (size_t)b * KK * DD + tn * 16;
  v8f dacc = {}, sacc = {};
  dacc = wmma_acc_rmaj(A_final + b * KK * KK + tm * 16 * KK, KK, Bb, DD, KK, dacc);
  sacc = wmma_acc_rmaj(A_static + tm * 16 * KK, KK, Bb, DD, KK, sacc);
  const int lane = threadIdx.x & 31, half = lane >> 4, l16 = lane & 15;
  float* dst = out + (size_t)b * KK * DD;
#pragma unroll
  for (int v = 0; v < 8; v++) {
    const float d = fmaxf(dacc[v], 0.f), s = fmaxf(sacc[v], 0.f);
    dst[(tm * 16 + v + 8 * half) * DD + tn * 16 + l16] = 0.5f * (d + s);
  }
}

extern "C" void kernel_launch(void* const* d_in, const int* in_sizes, int n_in,
                              void* d_out, int out_size, void* d_ws, size_t ws_size,
                              hipStream_t stream) {
  (void)in_sizes; (void)n_in; (void)out_size; (void)ws_size;
  const float* Adj    = (const float*)d_in[0];
  const float* nemb   = (const float*)d_in[1];
  const float* wfeat  = (const float*)d_in[2];
  const int*   mask   = (const int*)  d_in[3];
  const float* ht     = (const float*)d_in[4];
  const float* W_topk = (const float*)d_in[5];
  const float* b_topk = (const float*)d_in[6];
  const float* W_map  = (const float*)d_in[7];
  const float* b_map  = (const float*)d_in[8];
  const float* W_init = (const float*)d_in[9];
  const float* b_init = (const float*)d_in[10];
  const float* g_map  = (const float*)d_in[11]; // gcn_init_mapping [1,D]
  const float* A_stat = (const float*)d_in[12];
  const float* evA    = (const float*)d_in[13];
  const float* Wz = (const float*)d_in[14]; const float* Uz = (const float*)d_in[15]; const float* bz = (const float*)d_in[16];
  const float* Wr = (const float*)d_in[17]; const float* Ur = (const float*)d_in[18]; const float* br = (const float*)d_in[19];
  const float* Wh = (const float*)d_in[20]; const float* Uh = (const float*)d_in[21]; const float* bh = (const float*)d_in[22];

  float* out = (float*)d_out;
  float* w = (float*)d_ws;
  float* tanh_init = w + W_TANH;
  float* norms     = w + W_NORM;
  float* scores    = w + W_SCORES;
  float* tkv       = w + W_TKV;
  int*   tki       = (int*)(w + W_TKI);
  float* node_sel  = w + W_NSEL;
  float* sel_gate  = w + W_SGATE;
  float* wlf       = w + W_WLF;
  float* attn      = w + W_ATTN;
  float* A_final   = w + W_AF;
  float* base      = w + W_BASE;

  float* scorer   = out + O_SCORER;
  float* policy   = out + O_POL;
  float* entropy  = out + O_ENT;

  k_scorer<<<dim3(4, 8, 2), 32, 0, stream>>>(ht, W_topk, b_topk, W_init, b_init,
                                             scorer, tanh_init);
  k_norm<<<BB, 32, 0, stream>>>(scorer, norms);
  k_scores<<<(BB * NN) / 8, 256, 0, stream>>>(nemb, scorer, norms, mask, scores);
  k_topk<<<BB, 256, 0, stream>>>(scores, policy, entropy, tkv, tki);
  k_gather<<<BB * KK, DD, 0, stream>>>(nemb, tki, tkv, node_sel, sel_gate);
  k_wlf<<<dim3(LL / 16, DD / 16, BB), 32, 0, stream>>>(wfeat, W_map, b_map, wlf);
  k_attn<<<dim3(KK / 16, LL / 16, BB), 32, 0, stream>>>(node_sel, wlf, attn);
  k_afinal<<<BB, 128, 0, stream>>>(Adj, evA, Wz, Uz, bz, Wr, Ur, br, Wh, Uh, bh,
                                   sel_gate, attn, tki, A_final);
  k_base<<<BB * KK, 32, 0, stream>>>(node_sel, tanh_init, g_map, base);
  k_out<<<dim3(KK / 16, DD / 16, BB), 32, 0, stream>>>(A_final, A_stat, base, out);
}